// CPL_Loss_all_support_88785563943214
// MI455X (gfx1250) — compile-verified
//
#include <hip/hip_runtime.h>
#include <math.h>

// ---------------- problem constants (match reference) ----------------
#define NWAY   10
#define QPER   100
#define KSHOT  20
#define DDIM   2048
#define MNEG   10
#define NQ     (NWAY * QPER)        // 1000
#define NS     (NWAY * KSHOT)       // 200
#define SVAL   ((NWAY - 1) * MNEG + 1)  // 91 samples per query
#define SPAD   96                   // padded to 6 tiles of 16
#define TGTCOL ((NWAY - 1) * MNEG)  // 90: the query itself
#define ROWS   32                   // 20 support rows padded to 2 tiles
#define EPSV   1e-8f

typedef __attribute__((ext_vector_type(2))) float v2f;
typedef __attribute__((ext_vector_type(8))) float v8f;

__device__ __constant__ int kStrides[8] = {3, 7, 9, 11, 13, 17, 19, 21}; // coprime to 100

__device__ __forceinline__ unsigned hash_u32(unsigned x) {
    x ^= x >> 16; x *= 0x7feb352du;
    x ^= x >> 15; x *= 0x846ca68bu;
    x ^= x >> 16; return x;
}

// ---------------- kernel 1: row L2 norms (queries then supports) ----------------
__global__ __launch_bounds__(256) void row_norms(const float* __restrict__ qry,
                                                 const float* __restrict__ sup,
                                                 float* __restrict__ qn,
                                                 float* __restrict__ sn) {
    const int b = blockIdx.x;                        // 0..1199
    const float* row = (b < NQ) ? (qry + (size_t)b * DDIM)
                                : (sup + (size_t)(b - NQ) * DDIM);
    float s = 0.f;
    for (int i = threadIdx.x; i < DDIM; i += 256) {
        float v = row[i];
        s += v * v;
    }
    __shared__ float red[256];
    red[threadIdx.x] = s;
    __syncthreads();
    for (int off = 128; off > 0; off >>= 1) {
        if (threadIdx.x < off) red[threadIdx.x] += red[threadIdx.x + off];
        __syncthreads();
    }
    if (threadIdx.x == 0) {
        float n = sqrtf(red[0]);
        if (b < NQ) qn[b] = n; else sn[b - NQ] = n;
    }
}

// ---------------- kernel 2: per-query 20x91 cosine GEMM (WMMA) + row LSE ----------------
__global__ __launch_bounds__(128) void cpl_main(const float* __restrict__ sup,
                                                const float* __restrict__ qry,
                                                const float* __restrict__ qnorm,
                                                const float* __restrict__ snorm,
                                                float* __restrict__ part) {
    __shared__ int   gidx[SPAD];
    __shared__ float gn[SPAD];
    __shared__ float Cmat[ROWS][SPAD];   // 32 x 96 f32 = 12 KB
    __shared__ float wsum[4];

    const int q    = blockIdx.x;          // query id 0..999
    const int cls  = q / QPER;            // its class (block-structured labels)
    const int tid  = threadIdx.x;
    const int wave = tid >> 5;            // 0..3 (wave32)
    const int lane = tid & 31;

    // ---- deterministic negative sampling: M distinct in-class indices per other class ----
    if (tid < SPAD) {
        int s = tid, g;
        if (s >= TGTCOL) {
            g = q;                                    // target + padding = query itself
        } else {
            int j = s / MNEG, m = s % MNEG;
            int c = j + (j >= cls);                   // class != cls
            unsigned h = hash_u32((unsigned)(q * 131 + c * 7919 + 12345));
            int start  = (int)(h % 100u);
            int stride = kStrides[(h >> 8) & 7];      // gcd(stride,100)==1 -> distinct
            g = c * QPER + (start + m * stride) % QPER;
        }
        gidx[s] = g;
        gn[s]   = qnorm[g];
    }
    __syncthreads();

    // ---- wave tile assignment: tr = 16-row strip, three 16-col tiles each ----
    const int tr  = wave >> 1;            // 0,1
    const int tcb = (wave & 1) * 3;       // 0 or 3
    const int hf  = lane >> 4;            // K-half select per ISA 16x4 layout
    const int lan = lane & 15;

    int arow = tr * 16 + lan;                                  // padded support row
    int srow = cls * KSHOT + (arow < KSHOT ? arow : KSHOT - 1);// clamp padding in-bounds
    const float* Ap = sup + (size_t)srow * DDIM + 2 * hf;

    const int s0 = tcb * 16 + lan, s1 = s0 + 16, s2 = s0 + 32;
    const float* B0 = qry + (size_t)gidx[s0] * DDIM + 2 * hf;
    const float* B1 = qry + (size_t)gidx[s1] * DDIM + 2 * hf;
    const float* B2 = qry + (size_t)gidx[s2] * DDIM + 2 * hf;

    v8f acc0 = {}, acc1 = {}, acc2 = {};
    #pragma unroll 4
    for (int k = 0; k < DDIM; k += 4) {
        v2f a  = *(const v2f*)(Ap + k);
        v2f b0 = *(const v2f*)(B0 + k);
        v2f b1 = *(const v2f*)(B1 + k);
        v2f b2 = *(const v2f*)(B2 + k);
        acc0 = __builtin_amdgcn_wmma_f32_16x16x4_f32(false, a, false, b0, (short)0, acc0, false, false);
        acc1 = __builtin_amdgcn_wmma_f32_16x16x4_f32(false, a, false, b1, (short)0, acc1, false, false);
        acc2 = __builtin_amdgcn_wmma_f32_16x16x4_f32(false, a, false, b2, (short)0, acc2, false, false);
    }

    // ---- cosine scaling + spill to LDS (C/D layout: row = r + 8*hf, col in lane) ----
    const float g0 = gn[s0], g1 = gn[s1], g2 = gn[s2];
    #pragma unroll
    for (int r = 0; r < 8; ++r) {
        int row   = tr * 16 + r + 8 * hf;                      // 0..31
        int srowc = cls * KSHOT + (row < KSHOT ? row : KSHOT - 1);
        float snv = snorm[srowc];
        Cmat[row][s0] = acc0[r] / fmaxf(snv * g0, EPSV);
        Cmat[row][s1] = acc1[r] / fmaxf(snv * g1, EPSV);
        Cmat[row][s2] = acc2[r] / fmaxf(snv * g2, EPSV);
    }
    __syncthreads();

    // ---- per-row log-sum-exp over 91 valid cols; NLL at target col ----
    float wacc = 0.f;
    for (int row = wave; row < KSHOT; row += 4) {
        float mx = -3.0e38f;
        for (int c = lane; c < SVAL; c += 32) mx = fmaxf(mx, Cmat[row][c]);
        #pragma unroll
        for (int o = 16; o > 0; o >>= 1) mx = fmaxf(mx, __shfl_xor(mx, o, 32));
        float se = 0.f;
        for (int c = lane; c < SVAL; c += 32) se += expf(Cmat[row][c] - mx);
        #pragma unroll
        for (int o = 16; o > 0; o >>= 1) se += __shfl_xor(se, o, 32);
        if (lane == 0) wacc += (mx + logf(se)) - Cmat[row][TGTCOL];
    }
    if (lane == 0) wsum[wave] = wacc;
    __syncthreads();
    if (tid == 0) part[q] = wsum[0] + wsum[1] + wsum[2] + wsum[3];
}

// ---------------- kernel 3: deterministic fixed-order final reduction ----------------
__global__ __launch_bounds__(32) void final_reduce(const float* __restrict__ part,
                                                   float* __restrict__ out) {
    const int lane = threadIdx.x;
    float s = 0.f;
    for (int i = lane; i < NQ; i += 32) s += part[i];
    #pragma unroll
    for (int o = 16; o > 0; o >>= 1) s += __shfl_xor(s, o, 32);
    if (lane == 0) out[0] = s * (1.0f / ((float)NQ * (float)KSHOT * (float)NQ));
}

// ---------------- launcher ----------------
extern "C" void kernel_launch(void* const* d_in, const int* in_sizes, int n_in,
                              void* d_out, int out_size, void* d_ws, size_t ws_size,
                              hipStream_t stream) {
    const float* sup = (const float*)d_in[0];   // (200, 2048)
    const float* qry = (const float*)d_in[1];   // (1000, 2048)
    // d_in[2]/d_in[3]: labels — block-structured by construction (as the reference assumes)
    (void)in_sizes; (void)n_in; (void)out_size; (void)ws_size;

    float* ws   = (float*)d_ws;
    float* qn   = ws;            // [0, 1000)
    float* sn   = ws + NQ;       // [1000, 1200)
    float* part = ws + NQ + NS;  // [1200, 2200)
    float* out  = (float*)d_out;

    hipLaunchKernelGGL(row_norms,    dim3(NQ + NS), dim3(256), 0, stream, qry, sup, qn, sn);
    hipLaunchKernelGGL(cpl_main,     dim3(NQ),      dim3(128), 0, stream, sup, qry, qn, sn, part);
    hipLaunchKernelGGL(final_reduce, dim3(1),       dim3(32),  0, stream, part, out);
}